// DBGNNLayer_16338055594019
// MI455X (gfx1250) — compile-verified
//
#include <hip/hip_runtime.h>

#define DIM 128

typedef __attribute__((ext_vector_type(2))) float v2f;
typedef __attribute__((ext_vector_type(8))) float v8f;

// ---------------------------------------------------------------- zero
__global__ __launch_bounds__(256) void zero_kernel(float* p, long long n) {
    long long i = (long long)blockIdx.x * blockDim.x + threadIdx.x;
    long long stride = (long long)gridDim.x * blockDim.x;
    for (; i < n; i += stride) p[i] = 0.0f;
}

// ---------------------------------------------------------------- weight transpose
// Wt[n*128 + k] = W[k*128 + n]
__global__ __launch_bounds__(256) void transpose128(const float* __restrict__ W,
                                                    float* __restrict__ Wt) {
    int idx = blockIdx.x * blockDim.x + threadIdx.x;   // 16384 threads
    int n = idx >> 7, k = idx & 127;
    Wt[idx] = W[k * DIM + n];
}

// Wt[n*128 + k] = A[k*128 + n] + B[k*128 + n]   (folds x@Wr_buys + x@Wr_tags)
__global__ __launch_bounds__(256) void transpose_add128(const float* __restrict__ A,
                                                        const float* __restrict__ B,
                                                        float* __restrict__ Wt) {
    int idx = blockIdx.x * blockDim.x + threadIdx.x;
    int n = idx >> 7, k = idx & 127;
    Wt[idx] = A[k * DIM + n] + B[k * DIM + n];
}

// ---------------------------------------------------------------- edge scatter
// One wave (32 lanes) per edge: lane l handles dims [4l, 4l+3].
// Accumulators are L2-resident (153MB < 192MB L2) so f32 atomics run at L2 speed.
__global__ __launch_bounds__(256) void scatter_kernel(const float* __restrict__ xsrc,
                                                      const int* __restrict__ ei, int E,
                                                      float* __restrict__ acc,
                                                      float* __restrict__ cnt) {
    int gid  = blockIdx.x * blockDim.x + threadIdx.x;
    int e    = gid >> 5;                  // wave-uniform
    int lane = threadIdx.x & 31;
    if (e >= E) return;
    int src = ei[e];
    int dst = ei[E + e];
    const float4 v = *((const float4*)(xsrc + (size_t)src * DIM) + lane);
    float* a = acc + (size_t)dst * DIM + lane * 4;
    atomicAdd(a + 0, v.x);
    atomicAdd(a + 1, v.y);
    atomicAdd(a + 2, v.z);
    atomicAdd(a + 3, v.w);
    if (lane == 0) atomicAdd(cnt + dst, 1.0f);
}

// ---------------------------------------------------------------- WMMA helpers
// f32 WMMA fragment fetch, per ISA 16x4 A / 4x16 B layout:
// lanes 0-15 hold K={kk,kk+1}, lanes 16-31 hold K={kk+2,kk+3}.
__device__ __forceinline__ v8f wmma_f32(v2f a, v2f b, v8f c) {
    return __builtin_amdgcn_wmma_f32_16x16x4_f32(false, a, false, b, (short)0, c,
                                                 false, false);
}

// ---------------------------------------------------------------- user finalize
// out_user = (acc_rev/cnt) @ Wl_rev + x_user @ Wr_rev + b_rev
// acc_rev aliases out (row-local in-place: stage -> barrier -> overwrite own rows).
__global__ __launch_bounds__(256) void finalize_user(const float* __restrict__ acc_rev,
                                                     const float* __restrict__ x_user,
                                                     const float* __restrict__ cnt_rev,
                                                     const float* __restrict__ Wlt,
                                                     const float* __restrict__ Wrt,
                                                     const float* __restrict__ bias,
                                                     float* __restrict__ out) {
    __shared__ float Am[16 * DIM];
    __shared__ float Ax[16 * DIM];
    const int tile = blockIdx.x;
    const int t    = threadIdx.x;
    #pragma unroll
    for (int i = 0; i < 8; ++i) {                // stage 16x128 tiles, fuse mean
        int idx = t + i * 256;
        int r = idx >> 7, col = idx & 127;
        int grow = tile * 16 + r;
        float s = 1.0f / fmaxf(cnt_rev[grow], 1.0f);
        Am[idx] = acc_rev[(size_t)grow * DIM + col] * s;
        Ax[idx] = x_user[(size_t)grow * DIM + col];
    }
    __syncthreads();
    const int w = t >> 5, lane = t & 31;
    const int hh = lane >> 4;                    // K-half select
    const int lr = lane & 15;
    const int ncol = w * 16 + lr;
    v8f c = {};
    for (int kk = 0; kk < DIM; kk += 4) {
        int kb = kk + hh * 2;
        v2f am = *(const v2f*)&Am[lr * DIM + kb];
        v2f ax = *(const v2f*)&Ax[lr * DIM + kb];
        v2f b1 = *(const v2f*)&Wlt[ncol * DIM + kb];
        v2f b2 = *(const v2f*)&Wrt[ncol * DIM + kb];
        c = wmma_f32(am, b1, c);
        c = wmma_f32(ax, b2, c);
    }
    const float bv = bias[ncol];
    const int rbase = tile * 16 + hh * 8;
    #pragma unroll
    for (int j = 0; j < 8; ++j)
        out[(size_t)(rbase + j) * DIM + ncol] = c[j] + bv;
}

// ---------------------------------------------------------------- item finalize
// out_item = 0.5*( mean_buys@Wl_b + mean_tags@Wl_t + x_item@(Wr_b+Wr_t) + b_b + b_t )
// acc_buys aliases out (row-local in-place, safe as above).
__global__ __launch_bounds__(256) void finalize_item(const float* __restrict__ acc_buys,
                                                     const float* __restrict__ acc_tags,
                                                     const float* __restrict__ x_item,
                                                     const float* __restrict__ cnt_buys,
                                                     const float* __restrict__ cnt_tags,
                                                     const float* __restrict__ Wlbt,
                                                     const float* __restrict__ Wltt,
                                                     const float* __restrict__ Wct,
                                                     const float* __restrict__ b_buys,
                                                     const float* __restrict__ b_tags,
                                                     float* __restrict__ out) {
    __shared__ float Ab[16 * DIM];
    __shared__ float At[16 * DIM];
    __shared__ float Ax[16 * DIM];
    const int tile = blockIdx.x;
    const int t    = threadIdx.x;
    #pragma unroll
    for (int i = 0; i < 8; ++i) {
        int idx = t + i * 256;
        int r = idx >> 7, col = idx & 127;
        int grow = tile * 16 + r;
        float sb = 1.0f / fmaxf(cnt_buys[grow], 1.0f);
        float st = 1.0f / fmaxf(cnt_tags[grow], 1.0f);
        Ab[idx] = acc_buys[(size_t)grow * DIM + col] * sb;
        At[idx] = acc_tags[(size_t)grow * DIM + col] * st;
        Ax[idx] = x_item[(size_t)grow * DIM + col];
    }
    __syncthreads();
    const int w = t >> 5, lane = t & 31;
    const int hh = lane >> 4;
    const int lr = lane & 15;
    const int ncol = w * 16 + lr;
    v8f c = {};
    for (int kk = 0; kk < DIM; kk += 4) {
        int kb = kk + hh * 2;
        v2f ab = *(const v2f*)&Ab[lr * DIM + kb];
        v2f at = *(const v2f*)&At[lr * DIM + kb];
        v2f ax = *(const v2f*)&Ax[lr * DIM + kb];
        v2f b1 = *(const v2f*)&Wlbt[ncol * DIM + kb];
        v2f b2 = *(const v2f*)&Wltt[ncol * DIM + kb];
        v2f b3 = *(const v2f*)&Wct[ncol * DIM + kb];
        c = wmma_f32(ab, b1, c);
        c = wmma_f32(at, b2, c);
        c = wmma_f32(ax, b3, c);
    }
    const float bv = 0.5f * (b_buys[ncol] + b_tags[ncol]);
    const int rbase = tile * 16 + hh * 8;
    #pragma unroll
    for (int j = 0; j < 8; ++j)
        out[(size_t)(rbase + j) * DIM + ncol] = 0.5f * c[j] + bv;
}

// ---------------------------------------------------------------- launch
extern "C" void kernel_launch(void* const* d_in, const int* in_sizes, int n_in,
                              void* d_out, int out_size, void* d_ws, size_t ws_size,
                              hipStream_t stream) {
    const float* x_user  = (const float*)d_in[0];
    const float* x_item  = (const float*)d_in[1];
    const float* x_tag   = (const float*)d_in[2];
    const int*   ei_buys = (const int*)d_in[3];
    const int*   ei_rev  = (const int*)d_in[4];
    const int*   ei_tags = (const int*)d_in[5];
    const float* Wl_buys = (const float*)d_in[6];
    const float* Wr_buys = (const float*)d_in[7];
    const float* b_buys  = (const float*)d_in[8];
    const float* Wl_rev  = (const float*)d_in[9];
    const float* Wr_rev  = (const float*)d_in[10];
    const float* b_rev   = (const float*)d_in[11];
    const float* Wl_tags = (const float*)d_in[12];
    const float* Wr_tags = (const float*)d_in[13];
    const float* b_tags  = (const float*)d_in[14];

    const int N_USER = in_sizes[0] / DIM;
    const int N_ITEM = in_sizes[1] / DIM;
    const int E_BUYS = in_sizes[3] / 2;
    const int E_REV  = in_sizes[4] / 2;
    const int E_TAGS = in_sizes[5] / 2;

    float* out_user = (float*)d_out;                       // also acc_rev
    float* out_item = out_user + (size_t)N_USER * DIM;     // also acc_buys

    // workspace layout (floats): acc_tags | cnt_buys | cnt_rev | cnt_tags | 5x Wt
    float* ws       = (float*)d_ws;
    float* acc_tags = ws;
    float* cnt_buys = acc_tags + (size_t)N_ITEM * DIM;
    float* cnt_rev  = cnt_buys + N_ITEM;
    float* cnt_tags = cnt_rev + N_USER;
    float* wt_lrev  = cnt_tags + N_ITEM;
    float* wt_rrev  = wt_lrev + DIM * DIM;
    float* wt_lbuys = wt_rrev + DIM * DIM;
    float* wt_ltags = wt_lbuys + DIM * DIM;
    float* wt_comb  = wt_ltags + DIM * DIM;                // (Wr_buys+Wr_tags)^T

    // 1) zero accumulators + counts (d_out regions double as accumulators)
    long long n_out = (long long)(N_USER + N_ITEM) * DIM;
    long long n_ws  = (long long)N_ITEM * DIM + N_ITEM + N_USER + N_ITEM;
    zero_kernel<<<4096, 256, 0, stream>>>(out_user, n_out);
    zero_kernel<<<4096, 256, 0, stream>>>(ws, n_ws);

    // 2) transpose weights (B operands become contiguous float2 per lane)
    transpose128<<<64, 256, 0, stream>>>(Wl_rev, wt_lrev);
    transpose128<<<64, 256, 0, stream>>>(Wr_rev, wt_rrev);
    transpose128<<<64, 256, 0, stream>>>(Wl_buys, wt_lbuys);
    transpose128<<<64, 256, 0, stream>>>(Wl_tags, wt_ltags);
    transpose_add128<<<64, 256, 0, stream>>>(Wr_buys, Wr_tags, wt_comb);

    // 3) edge scatter-accumulate (atomics into L2-resident accumulators)
    scatter_kernel<<<E_BUYS / 8, 256, 0, stream>>>(x_user, ei_buys, E_BUYS,
                                                   out_item, cnt_buys);
    scatter_kernel<<<E_REV / 8, 256, 0, stream>>>(x_item, ei_rev, E_REV,
                                                  out_user, cnt_rev);
    scatter_kernel<<<E_TAGS / 8, 256, 0, stream>>>(x_tag, ei_tags, E_TAGS,
                                                   acc_tags, cnt_tags);

    // 4) fused mean + WMMA GEMMs + bias (+ 0.5 cross-relation mean for items)
    finalize_user<<<N_USER / 16, 256, 0, stream>>>(out_user, x_user, cnt_rev,
                                                   wt_lrev, wt_rrev, b_rev, out_user);
    finalize_item<<<N_ITEM / 16, 256, 0, stream>>>(out_item, acc_tags, x_item,
                                                   cnt_buys, cnt_tags,
                                                   wt_lbuys, wt_ltags, wt_comb,
                                                   b_buys, b_tags, out_item);
}